// ResidualFourierBlock1D_38439957299540
// MI455X (gfx1250) — compile-verified
//
#include <hip/hip_runtime.h>

// ---------------------------------------------------------------------------
// FNO residual block, MI455X (gfx1250), f32 WMMA (V_WMMA_F32_16X16X4_F32).
//
// spectral_conv keeps only 33 rfft modes of 8192 -> replace FFT with dense
// cos/sin-basis GEMMs:
//   G[b](80x64)  = F(80x8192) @ x[b]           (rows 0..32 cos_k, 33..65 sin_k, 66..79 zero)
//   Yhat[b]      = complex per-mode 64x64 mix  (rows 0..32 Yr, 33..65 Yi, 66..79 zero)
//   S[b](8192x64)= Ft(8192x80) @ Yhat[b]       (1/N, 2x factors folded into Ft; sin0 col = 0
//                                               so Im(Y0) is ignored, matching irfft)
// iDFT kernels compute all 4 column tiles per wave so A-fragments (Ft, x) are
// loaded once and reused across 4 WMMAs per K-step.
// ---------------------------------------------------------------------------

#define BATCH 32
#define NPTS  8192
#define CH    64
#define K33   33
#define R80   80          // padded basis rows (66 used)
#define EDIM  128
#define SPLITS 8          // split-K chunks over n for forward DFT

typedef float v2f __attribute__((ext_vector_type(2)));
typedef float v8f __attribute__((ext_vector_type(8)));

// ---------------------------------------------------------------- basis tables
__global__ void build_F(float* __restrict__ F) {
  int id = blockIdx.x * blockDim.x + threadIdx.x;      // 80*8192 threads
  if (id >= R80 * NPTS) return;
  int n = id & (NPTS - 1);
  int r = id >> 13;
  float v = 0.0f;
  const double step = 6.283185307179586 / (double)NPTS;
  if (r < K33) {
    int m = (r * n) & (NPTS - 1);
    v = (float)cos((double)m * step);
  } else if (r < 2 * K33) {
    int k = r - K33;
    int m = (k * n) & (NPTS - 1);
    v = (float)sin((double)m * step);
  }
  F[id] = v;
}

__global__ void build_Ft(float* __restrict__ Ft) {
  int id = blockIdx.x * blockDim.x + threadIdx.x;      // 8192*80 threads
  if (id >= NPTS * R80) return;
  int r = id % R80;
  int n = id / R80;
  const double step = 6.283185307179586 / (double)NPTS;
  const float inv = 1.0f / (float)NPTS;
  float v = 0.0f;
  if (r == 0) {
    v = inv;                                           // k=0 cos term, weight 1/N
  } else if (r < K33) {
    int m = (r * n) & (NPTS - 1);
    v = 2.0f * inv * (float)cos((double)m * step);
  } else if (r > K33 && r < 2 * K33) {                 // r==33 (sin_0) stays 0 -> drops Im(Y0)
    int k = r - K33;
    int m = (k * n) & (NPTS - 1);
    v = -2.0f * inv * (float)sin((double)m * step);
  }
  Ft[id] = v;
}

// ---------------------------------------------------------------- time MLP
__global__ void time_mlp(const float* __restrict__ t_emb, const float* __restrict__ tk,
                         const float* __restrict__ tb, float* __restrict__ tm) {
  int id = blockIdx.x * blockDim.x + threadIdx.x;      // 32*128 threads
  if (id >= BATCH * EDIM) return;
  int j = id & (EDIM - 1);
  int b = id >> 7;
  float acc = tb[j];
  for (int e = 0; e < EDIM; ++e)
    acc = fmaf(t_emb[b * EDIM + e], tk[e * EDIM + j], acc);
  tm[id] = acc;
}

// ------------------------------------------------- forward DFT: P[b,s] = F@in chunk
// one 32-lane wave per (b, coltile, split); 5 row-tiles of 16 -> 80 rows
__global__ void dft_gemm(const float* __restrict__ F, const float* __restrict__ X,
                         float* __restrict__ P) {
  int wid  = blockIdx.x * 4 + (threadIdx.x >> 5);      // 1024 waves total
  int lane = threadIdx.x & 31;
  int s  = wid & (SPLITS - 1);
  int ct = (wid >> 3) & 3;
  int b  = wid >> 5;
  int l15 = lane & 15;
  int hi2 = (lane >= 16) ? 2 : 0;
  const int KC = NPTS / SPLITS;                        // 1024
  int k0base = s * KC;

  const float* xb = X + (size_t)b * NPTS * CH;
  int col = ct * 16 + l15;

  v8f acc[5];
  #pragma unroll
  for (int rt = 0; rt < 5; ++rt) acc[rt] = (v8f){0.f,0.f,0.f,0.f,0.f,0.f,0.f,0.f};

  for (int k0 = k0base; k0 < k0base + KC; k0 += 4) {
    int ka = k0 + hi2;
    v2f bf;
    bf.x = xb[(size_t)ka * CH + col];
    bf.y = xb[(size_t)(ka + 1) * CH + col];
    #pragma unroll
    for (int rt = 0; rt < 5; ++rt) {
      int row = rt * 16 + l15;
      v2f af = *(const v2f*)(F + (size_t)row * NPTS + ka);
      acc[rt] = __builtin_amdgcn_wmma_f32_16x16x4_f32(false, af, false, bf,
                                                      (short)0, acc[rt], false, false);
    }
  }

  float* pb = P + ((size_t)b * SPLITS + s) * (R80 * CH);
  int rbase = (lane >= 16) ? 8 : 0;
  #pragma unroll
  for (int rt = 0; rt < 5; ++rt)
    #pragma unroll
    for (int j = 0; j < 8; ++j)
      pb[(rt * 16 + j + rbase) * CH + col] = acc[rt][j];
}

// ------------------------------------------------- reduce split-K partials
__global__ void reduce_p(const float* __restrict__ P, float* __restrict__ G) {
  int id = blockIdx.x * blockDim.x + threadIdx.x;      // 32*80*64 threads
  if (id >= BATCH * R80 * CH) return;
  int rc = id % (R80 * CH);
  int b  = id / (R80 * CH);
  float a = 0.0f;
  for (int s = 0; s < SPLITS; ++s)
    a += P[((size_t)b * SPLITS + s) * (R80 * CH) + rc];
  G[(size_t)b * (R80 * CH) + rc] = a;
}

// ------------------------------------------------- per-mode complex mix
// G rows: 0..32 = Xr, 33..65 = xs (= Sum x*sin, so Xi = -xs). Y rows: Yr / Yi / zeros.
__global__ void modemix(const float* __restrict__ G, const float* __restrict__ Wr,
                        const float* __restrict__ Wi, float* __restrict__ Y) {
  int id = blockIdx.x * blockDim.x + threadIdx.x;      // 32*47*64 threads exactly
  int o = id & 63;
  int k = (id >> 6) % 47;
  int b = id / (47 * 64);
  float* yb = Y + (size_t)b * (R80 * CH);
  if (k >= K33) {                                      // zero pad rows 66..79
    yb[(K33 + k) * CH + o] = 0.0f;
    return;
  }
  const float* gb = G + (size_t)b * (R80 * CH);
  float yr = 0.0f, yi = 0.0f;
  for (int i = 0; i < CH; ++i) {
    float xr = gb[k * CH + i];
    float xs = gb[(K33 + k) * CH + i];
    float wr = Wr[((size_t)k * CH + i) * CH + o];
    float wi = Wi[((size_t)k * CH + i) * CH + o];
    yr = fmaf(xr, wr, fmaf(xs, wi, yr));               // Yr = Xr*Wr - Xi*Wi, Xi = -xs
    yi = fmaf(xr, wi, fmaf(-xs, wr, yi));              // Yi = Xr*Wi + Xi*Wr
  }
  yb[k * CH + o] = yr;
  yb[(K33 + k) * CH + o] = yi;
}

// ------------------------------------------------- iDFT1 + relu + FiLM + relu -> A2
// one wave per (b, ntile): 16 rows x all 64 cols; A-frag (Ft) shared across 4 col tiles
__global__ void idft_film(const float* __restrict__ Ft, const float* __restrict__ Y,
                          const float* __restrict__ tm, float* __restrict__ A2) {
  int wid  = blockIdx.x * 4 + (threadIdx.x >> 5);      // 16384 waves
  int lane = threadIdx.x & 31;
  int nt = wid & 511;
  int b  = wid >> 9;
  int l15 = lane & 15;
  int hi2 = (lane >= 16) ? 2 : 0;
  int n0  = nt * 16;

  const float* yb  = Y + (size_t)b * (R80 * CH);
  const float* fta = Ft + (size_t)(n0 + l15) * R80 + hi2;

  v8f acc[4];
  #pragma unroll
  for (int ct = 0; ct < 4; ++ct) acc[ct] = (v8f){0.f,0.f,0.f,0.f,0.f,0.f,0.f,0.f};

  #pragma unroll
  for (int k0 = 0; k0 < R80; k0 += 4) {
    v2f af = *(const v2f*)(fta + k0);
    int ka = k0 + hi2;
    #pragma unroll
    for (int ct = 0; ct < 4; ++ct) {
      int col = ct * 16 + l15;
      v2f bf;
      bf.x = yb[ka * CH + col];
      bf.y = yb[(ka + 1) * CH + col];
      acc[ct] = __builtin_amdgcn_wmma_f32_16x16x4_f32(false, af, false, bf,
                                                      (short)0, acc[ct], false, false);
    }
  }

  float* a2b = A2 + ((size_t)b * NPTS + n0) * CH;
  int rbase = (lane >= 16) ? 8 : 0;
  #pragma unroll
  for (int ct = 0; ct < 4; ++ct) {
    int col = ct * 16 + l15;
    float w  = tm[b * EDIM + col] + 1.0f;
    float bb = tm[b * EDIM + CH + col];
    #pragma unroll
    for (int j = 0; j < 8; ++j) {
      float v = fmaxf(acc[ct][j], 0.0f);
      v = fmaf(v, w, bb);
      v = fmaxf(v, 0.0f);
      a2b[(j + rbase) * CH + col] = v;
    }
  }
}

// ------------------------------------------------- iDFT2 + x@res_kernel + bias -> out
// one wave per (b, ntile): A-frags (Ft and x rows) shared across 4 col tiles
__global__ void final_k(const float* __restrict__ Ft, const float* __restrict__ Y,
                        const float* __restrict__ X, const float* __restrict__ Wres,
                        const float* __restrict__ bres, float* __restrict__ out) {
  int wid  = blockIdx.x * 4 + (threadIdx.x >> 5);      // 16384 waves
  int lane = threadIdx.x & 31;
  int nt = wid & 511;
  int b  = wid >> 9;
  int l15 = lane & 15;
  int hi2 = (lane >= 16) ? 2 : 0;
  int n0  = nt * 16;

  const float* yb  = Y + (size_t)b * (R80 * CH);
  const float* fta = Ft + (size_t)(n0 + l15) * R80 + hi2;

  v8f acc[4];
  #pragma unroll
  for (int ct = 0; ct < 4; ++ct) acc[ct] = (v8f){0.f,0.f,0.f,0.f,0.f,0.f,0.f,0.f};

  #pragma unroll
  for (int k0 = 0; k0 < R80; k0 += 4) {
    v2f af = *(const v2f*)(fta + k0);
    int ka = k0 + hi2;
    #pragma unroll
    for (int ct = 0; ct < 4; ++ct) {
      int col = ct * 16 + l15;
      v2f bf;
      bf.x = yb[ka * CH + col];
      bf.y = yb[(ka + 1) * CH + col];
      acc[ct] = __builtin_amdgcn_wmma_f32_16x16x4_f32(false, af, false, bf,
                                                      (short)0, acc[ct], false, false);
    }
  }

  // residual x @ res_kernel
  const float* xrow = X + ((size_t)b * NPTS + n0 + l15) * CH + hi2;
  #pragma unroll
  for (int k0 = 0; k0 < CH; k0 += 4) {
    v2f af = *(const v2f*)(xrow + k0);
    int ka = k0 + hi2;
    #pragma unroll
    for (int ct = 0; ct < 4; ++ct) {
      int col = ct * 16 + l15;
      v2f bf;
      bf.x = Wres[ka * CH + col];
      bf.y = Wres[(ka + 1) * CH + col];
      acc[ct] = __builtin_amdgcn_wmma_f32_16x16x4_f32(false, af, false, bf,
                                                      (short)0, acc[ct], false, false);
    }
  }

  float* ob = out + ((size_t)b * NPTS + n0) * CH;
  int rbase = (lane >= 16) ? 8 : 0;
  #pragma unroll
  for (int ct = 0; ct < 4; ++ct) {
    int col = ct * 16 + l15;
    float bias = bres[col];
    #pragma unroll
    for (int j = 0; j < 8; ++j)
      ob[(j + rbase) * CH + col] = acc[ct][j] + bias;
  }
}

// ---------------------------------------------------------------------------
extern "C" void kernel_launch(void* const* d_in, const int* in_sizes, int n_in,
                              void* d_out, int out_size, void* d_ws, size_t ws_size,
                              hipStream_t stream) {
  const float* x    = (const float*)d_in[0];
  const float* temb = (const float*)d_in[1];
  const float* w1r  = (const float*)d_in[2];
  const float* w1i  = (const float*)d_in[3];
  const float* w2r  = (const float*)d_in[4];
  const float* w2i  = (const float*)d_in[5];
  const float* tk   = (const float*)d_in[6];
  const float* tb   = (const float*)d_in[7];
  const float* resk = (const float*)d_in[8];
  const float* resb = (const float*)d_in[9];
  float* out = (float*)d_out;
  float* ws  = (float*)d_ws;

  // workspace layout (floats)
  float* F  = ws;                                   // 80*8192
  float* Ft = F  + (size_t)R80 * NPTS;              // 8192*80
  float* tm = Ft + (size_t)NPTS * R80;              // 32*128
  float* P  = tm + (size_t)BATCH * EDIM;            // 32*8*80*64
  float* G  = P  + (size_t)BATCH * SPLITS * R80 * CH;
  float* Y  = G  + (size_t)BATCH * R80 * CH;
  float* A2 = Y  + (size_t)BATCH * R80 * CH;        // 32*8192*64

  build_F <<<(R80 * NPTS + 255) / 256, 256, 0, stream>>>(F);
  build_Ft<<<(NPTS * R80 + 255) / 256, 256, 0, stream>>>(Ft);
  time_mlp<<<(BATCH * EDIM + 255) / 256, 256, 0, stream>>>(temb, tk, tb, tm);

  // spectral conv 1
  dft_gemm<<<(BATCH * 4 * SPLITS) / 4, 128, 0, stream>>>(F, x, P);
  reduce_p<<<(BATCH * R80 * CH + 255) / 256, 256, 0, stream>>>(P, G);
  modemix <<<(BATCH * 47 * CH) / 256, 256, 0, stream>>>(G, w1r, w1i, Y);
  idft_film<<<(BATCH * 512) / 4, 128, 0, stream>>>(Ft, Y, tm, A2);

  // spectral conv 2
  dft_gemm<<<(BATCH * 4 * SPLITS) / 4, 128, 0, stream>>>(F, A2, P);
  reduce_p<<<(BATCH * R80 * CH + 255) / 256, 256, 0, stream>>>(P, G);
  modemix <<<(BATCH * 47 * CH) / 256, 256, 0, stream>>>(G, w2r, w2i, Y);

  // iDFT2 + residual GEMM + bias
  final_k<<<(BATCH * 512) / 4, 128, 0, stream>>>(Ft, Y, x, resk, resb, out);
}